// ECLReweightedLoss_8366596293206
// MI455X (gfx1250) — compile-verified
//
#include <hip/hip_runtime.h>
#include <hip/hip_bf16.h>
#include <stdint.h>

// ECL reweighted supervised-contrastive loss, fused.
// N=8192 rows, D=128. sim = F F^T / tau via V_WMMA_F32_16X16X4_F32.
// Revision: column loop split over 4 slices (512 blocks / 2048 waves for
// occupancy-based latency hiding), online-softmax partials merged by a
// deterministic single-block reduction (no float atomics). Staging uses
// GLOBAL_LOAD_ASYNC_TO_LDS_B128 with ASYNCcnt ping-pong.

#define N_TOT   8192
#define D_DIM   128
#define T_SEQ   128
#define K_PROTO 16
#define TAU_F   0.07f

#define WAVES_PER_BLOCK 4
#define ROWS_PER_WAVE   16
#define ROWS_PER_BLOCK  (WAVES_PER_BLOCK * ROWS_PER_WAVE)
#define N_TILES         (N_TOT / 16)
#define COL_SLICES      4
#define TILES_PER_SLICE (N_TILES / COL_SLICES)
#define TILE_ROW_F2     34                       // float2 row stride: 272B (16B-aligned, conflict-free)
#define TILE_F2         (16 * TILE_ROW_F2)
#define TILE_BYTES      (TILE_F2 * 8)

// Workspace layout (float offsets)
#define WS_PM   0                       // [COL_SLICES][N_TOT] running max
#define WS_PS   (COL_SLICES * N_TOT)    // [COL_SLICES][N_TOT] exp-sum (rel. to max)
#define WS_PMS  (2 * COL_SLICES * N_TOT)// [COL_SLICES][N_TOT] masked sim-sum
#define WS_PC   (3 * COL_SLICES * N_TOT)// [COL_SLICES][N_TOT] matching-tile count
#define WS_NRM  (4 * COL_SLICES * N_TOT)// [N_TOT] |f|^2
#define WS_PRT  (WS_NRM + N_TOT)        // [N_TOT] f . pbar
#define WS_PBAR (WS_PRT + N_TOT)        // [D_DIM] mean prototype

typedef float v2f __attribute__((ext_vector_type(2)));
typedef float v8f __attribute__((ext_vector_type(8)));

// ---------------------------------------------------------------------------
// Prep: pbar[d] = mean_k prototypes[k][d].
// ---------------------------------------------------------------------------
__global__ void ecl_prep_kernel(const float* __restrict__ protos,
                                float* __restrict__ ws) {
  int d = threadIdx.x;
  if (d < D_DIM) {
    float s = 0.f;
    #pragma unroll
    for (int k = 0; k < K_PROTO; ++k) s += protos[k * D_DIM + d];
    ws[WS_PBAR + d] = s * (1.0f / (float)K_PROTO);
  }
}

// Async global -> LDS, 16 bytes per lane, tracked by ASYNCcnt.
__device__ __forceinline__ void async_to_lds_b128(unsigned lds_addr,
                                                  int gvoff_bytes,
                                                  const float* gbase) {
  asm volatile("global_load_async_to_lds_b128 %0, %1, %2"
               :: "v"(lds_addr), "v"(gvoff_bytes), "s"(gbase)
               : "memory");
}
__device__ __forceinline__ void wait_asynccnt0() {
  asm volatile("s_wait_asynccnt 0" ::: "memory");
}

// ---------------------------------------------------------------------------
// Main kernel: per (64-row block, 2048-column slice) online-softmax partials.
// ---------------------------------------------------------------------------
__global__ __launch_bounds__(WAVES_PER_BLOCK * 32, 1)
void ecl_main_kernel(const float* __restrict__ feat,
                     const int*   __restrict__ labels,
                     float* __restrict__ ws) {
  __shared__ float2 tile[2][TILE_F2];                     // ping-pong 16x128 col tile

  const int tid   = threadIdx.x;
  const int wave  = tid >> 5;
  const int lane  = tid & 31;
  const int l16   = lane & 15;
  const int hi    = lane >> 4;                             // lane-half: K offset +2
  const int slice = blockIdx.y;

  const int   i0     = blockIdx.x * ROWS_PER_BLOCK + wave * ROWS_PER_WAVE;
  const float invtau = 1.0f / TAU_F;
  const float* pbar  = ws + WS_PBAR;

  // Per-thread async-staging addresses: 4x b128 (one float4 each) per tile.
  unsigned dst_off[4];
  int      gvoff[4];
  {
    const unsigned ldsbase = (unsigned)(uintptr_t)&tile[0][0];
    #pragma unroll
    for (int it = 0; it < 4; ++it) {
      const int f  = tid + it * (WAVES_PER_BLOCK * 32);    // 0..511 float4 index
      const int r  = f >> 5, c4 = f & 31;
      dst_off[it]  = ldsbase + (unsigned)(r * (TILE_ROW_F2 * 8) + c4 * 16);
      gvoff[it]    = f * 16;
    }
  }

  // ---- A panel: 16 rows x 128 dims as 32 f32 16x16x4 A fragments.
  v2f a[32];
  {
    const float2* rowp = (const float2*)(feat + (size_t)(i0 + l16) * D_DIM);
    #pragma unroll
    for (int ks = 0; ks < 32; ++ks) {
      float2 v = rowp[2 * ks + hi];
      a[ks].x = v.x; a[ks].y = v.y;
    }
  }

  // ---- per-row |f|^2 and f . pbar, written once (slice 0).
  if (slice == 0) {
    float np = 0.f, dp = 0.f;
    #pragma unroll
    for (int ks = 0; ks < 32; ++ks) {
      const int koff = 4 * ks + 2 * hi;
      np += a[ks].x * a[ks].x + a[ks].y * a[ks].y;
      dp += a[ks].x * pbar[koff] + a[ks].y * pbar[koff + 1];
    }
    np += __shfl_xor(np, 16);
    dp += __shfl_xor(dp, 16);
    if (hi == 0) {
      ws[WS_NRM + i0 + l16] = np;
      ws[WS_PRT + i0 + l16] = dp;
    }
  }

  const int lbl_i = labels[i0 / T_SEQ];

  // Online-softmax state per C slot (slot p: row p or p+8, col class = lane%16).
  float m[8], s[8], ms[8];
  #pragma unroll
  for (int p = 0; p < 8; ++p) { m[p] = -1.0e30f; s[p] = 0.f; ms[p] = 0.f; }
  int nmatch = 0;

  const int jt0 = slice * TILES_PER_SLICE;
  const int jt1 = jt0 + TILES_PER_SLICE;

  // ---- prologue: async-stage first column tile of this slice into buffer 0.
  #pragma unroll
  for (int it = 0; it < 4; ++it)
    async_to_lds_b128(dst_off[it], gvoff[it], feat + (size_t)jt0 * 16 * D_DIM);

  for (int jt = jt0; jt < jt1; ++jt) {
    const int cur = (jt - jt0) & 1;
    // Own async writes into tile[cur] done; barrier makes all writes visible
    // and guarantees everyone finished reading tile[1-cur].
    wait_asynccnt0();
    __syncthreads();

    if (jt + 1 < jt1) {
      const float* gbase = feat + (size_t)(jt + 1) * 16 * D_DIM;
      const unsigned buf = (unsigned)((1 - cur) * TILE_BYTES);
      #pragma unroll
      for (int it = 0; it < 4; ++it)
        async_to_lds_b128(dst_off[it] + buf, gvoff[it], gbase);
    }

    const float2* tl = &tile[cur][0];

    // B fragments: 2-deep pipeline in groups of 4.
    v2f bb[2][4];
    #pragma unroll
    for (int q = 0; q < 4; ++q) {
      float2 t = tl[l16 * TILE_ROW_F2 + 2 * q + hi];
      bb[0][q].x = t.x; bb[0][q].y = t.y;
    }

    v8f c0 = {0.f, 0.f, 0.f, 0.f, 0.f, 0.f, 0.f, 0.f};
    v8f c1 = {0.f, 0.f, 0.f, 0.f, 0.f, 0.f, 0.f, 0.f};
    #pragma unroll
    for (int g = 0; g < 8; ++g) {
      const int pb = g & 1;
      if (g < 7) {
        #pragma unroll
        for (int q = 0; q < 4; ++q) {
          float2 t = tl[l16 * TILE_ROW_F2 + 2 * (4 * (g + 1) + q) + hi];
          bb[pb ^ 1][q].x = t.x; bb[pb ^ 1][q].y = t.y;
        }
      }
      c0 = __builtin_amdgcn_wmma_f32_16x16x4_f32(false, a[4 * g + 0], false, bb[pb][0],
                                                 (short)0, c0, false, false);
      c1 = __builtin_amdgcn_wmma_f32_16x16x4_f32(false, a[4 * g + 1], false, bb[pb][1],
                                                 (short)0, c1, false, false);
      c0 = __builtin_amdgcn_wmma_f32_16x16x4_f32(false, a[4 * g + 2], false, bb[pb][2],
                                                 (short)0, c0, false, false);
      c1 = __builtin_amdgcn_wmma_f32_16x16x4_f32(false, a[4 * g + 3], false, bb[pb][3],
                                                 (short)0, c1, false, false);
    }

    // Positive mask is uniform per tile (16 | T).
    const int   match = (labels[jt / (T_SEQ / 16)] == lbl_i);
    const float fm    = (float)match;
    nmatch += match;

    // Online softmax: single exp per element.
    #pragma unroll
    for (int p = 0; p < 8; ++p) {
      const float simv = (c0[p] + c1[p]) * invtau;
      const float d    = m[p] - simv;                    // >=0 if max unchanged
      const float e    = __expf(-fabsf(d));
      const float sa   = s[p] + e;                       // max unchanged
      const float sb   = __builtin_fmaf(s[p], e, 1.0f);  // new max: rescale
      s[p]  = (d >= 0.f) ? sa : sb;
      m[p]  = fmaxf(m[p], simv);
      ms[p] += fm * simv;
    }
  }

  // Merge the 16 column classes of each row (xor tree within each lane half).
  #pragma unroll
  for (int p = 0; p < 8; ++p) {
    #pragma unroll
    for (int msk = 1; msk < 16; msk <<= 1) {
      const float mo  = __shfl_xor(m[p],  msk);
      const float so  = __shfl_xor(s[p],  msk);
      const float mso = __shfl_xor(ms[p], msk);
      const float nm  = fmaxf(m[p], mo);
      s[p]  = s[p] * __expf(m[p] - nm) + so * __expf(mo - nm);
      ms[p] = ms[p] + mso;
      m[p]  = nm;
    }
  }

  // Lanes 0 (rows p) and 16 (rows p+8) store partials for 8 rows each.
  if (l16 == 0) {
    #pragma unroll
    for (int p = 0; p < 8; ++p) {
      const int row = i0 + p + 8 * hi;
      const int idx = slice * N_TOT + row;
      ws[WS_PM  + idx] = m[p];
      ws[WS_PS  + idx] = s[p];
      ws[WS_PMS + idx] = ms[p];
      ws[WS_PC  + idx] = (float)nmatch;
    }
  }
}

// ---------------------------------------------------------------------------
// Deterministic finalize: merge slice partials per row, epilogue, tree-sum.
// Single block, 256 threads; no atomics.
// ---------------------------------------------------------------------------
__global__ __launch_bounds__(256, 1)
void ecl_reduce_kernel(const int* __restrict__ labels,
                       const float* __restrict__ ws,
                       float* __restrict__ out) {
  __shared__ float red[256];
  const int tid = threadIdx.x;
  const float invtau = 1.0f / TAU_F;

  float acc = 0.f;
  for (int r = tid; r < N_TOT; r += 256) {
    float m  = ws[WS_PM + r];
    float s  = ws[WS_PS + r];
    float ms = ws[WS_PMS + r];
    float pc = ws[WS_PC + r];
    #pragma unroll
    for (int sl = 1; sl < COL_SLICES; ++sl) {
      const int idx = sl * N_TOT + r;
      const float mo  = ws[WS_PM + idx];
      const float so  = ws[WS_PS + idx];
      const float nm  = fmaxf(m, mo);
      s  = s * __expf(m - nm) + so * __expf(mo - nm);
      ms += ws[WS_PMS + idx];
      pc += ws[WS_PC + idx];
      m   = nm;
    }
    const float sim_self = ws[WS_NRM + r] * invtau;
    const float s_adj    = s - __expf(sim_self - m);    // drop self from exp-sum
    const float ms_adj   = ms - sim_self;               // drop self from pos-sum
    const float pos      = 16.0f * pc - 1.0f;           // exact positive count
    const float posc     = fmaxf(pos, 1.0f);
    const float denom    = s_adj + 1e-8f;
    float mlpp = (ms_adj - pos * m - pos * __logf(denom)) / posc;

    // adaptive re-weighting of fake rows (label == 1)
    const float avg   = ws[WS_PRT + r];                 // = mean_k f.proto_k
    const float avgc  = fminf(fmaxf(avg, 0.f), 1.f);
    const float wbase = 1.f - avgc;
    const float wgt   = fmaxf(wbase * sqrtf(wbase), 0.05f);  // wbase^1.5
    if (labels[r / T_SEQ] == 1) mlpp *= wgt;

    acc += -mlpp;                                       // -(TAU/TAU0) == -1
  }

  red[tid] = acc;
  __syncthreads();
  #pragma unroll
  for (int st = 128; st > 0; st >>= 1) {
    if (tid < st) red[tid] += red[tid + st];
    __syncthreads();
  }
  if (tid == 0) out[0] = red[0] * (1.0f / (float)N_TOT);
}

// ---------------------------------------------------------------------------
extern "C" void kernel_launch(void* const* d_in, const int* in_sizes, int n_in,
                              void* d_out, int out_size, void* d_ws, size_t ws_size,
                              hipStream_t stream) {
  const float* feat   = (const float*)d_in[0];
  const int*   labels = (const int*)d_in[1];
  const float* protos = (const float*)d_in[2];
  float*       out    = (float*)d_out;
  float*       ws     = (float*)d_ws;

  ecl_prep_kernel<<<1, 128, 0, stream>>>(protos, ws);
  dim3 grid(N_TOT / ROWS_PER_BLOCK, COL_SLICES);
  ecl_main_kernel<<<grid, WAVES_PER_BLOCK * 32, 0, stream>>>(feat, labels, ws);
  ecl_reduce_kernel<<<1, 256, 0, stream>>>(labels, ws, out);
}